// Net_31001073942546
// MI455X (gfx1250) — compile-verified
//
#include <hip/hip_runtime.h>

// ---------------------------------------------------------------------------
// instant-NGP hash encode (L=4, F=2, T=19) + MLP 8->16->16->2, fused.
// WMMA (V_WMMA_F32_16X16X4_F32) for the two 16-wide layers, per-wave LDS
// staging, wave-local ordering only (LDS is in-order per wave on CDNA5).
// ---------------------------------------------------------------------------

typedef float v2f __attribute__((ext_vector_type(2)));
typedef float v8f __attribute__((ext_vector_type(8)));

#define TBITS 19
#define TSIZE (1u << TBITS)
#define HMASK (TSIZE - 1u)
#define P1 2654435761u
#define P2 805459861u
#define WPB 8   // waves per 256-thread block

__device__ __forceinline__ void lds_fence_wave() {
  // Order LDS store-phase vs load-phase for this wave. Hardware keeps DS ops
  // in-order per wave; the asm memory clobber stops compiler reordering.
  asm volatile("s_wait_dscnt 0x0" ::: "memory");
}

__device__ __forceinline__ void encode_level(float sx, float sy, float sz,
                                             const float* __restrict__ tl,
                                             float& o0, float& o1) {
  float fx = floorf(sx), fy = floorf(sy), fz = floorf(sz);
  float wx = sx - fx, wy = sy - fy, wz = sz - fz;
  float wx0 = 1.f - wx, wy0 = 1.f - wy, wz0 = 1.f - wz;
  unsigned ix = (unsigned)(int)fx, iy = (unsigned)(int)fy, iz = (unsigned)(int)fz;
  unsigned ux0 = ix,        ux1 = ix + 1u;
  unsigned uy0 = iy * P1,   uy1 = (iy + 1u) * P1;
  unsigned uz0 = iz * P2,   uz1 = (iz + 1u) * P2;
  float a0 = 0.f, a1 = 0.f;
#pragma unroll
  for (int c = 0; c < 8; ++c) {
    unsigned h = ((c & 1) ? ux1 : ux0) ^ ((c & 2) ? uy1 : uy0) ^ ((c & 4) ? uz1 : uz0);
    float wt = ((c & 1) ? wx : wx0) * ((c & 2) ? wy : wy0) * ((c & 4) ? wz : wz0);
    const float2 tv = *(const float2*)(tl + 2u * (h & HMASK));  // global_load_b64, L2-resident
    a0 = fmaf(wt, tv.x, a0);
    a1 = fmaf(wt, tv.y, a1);
  }
  o0 = a0; o1 = a1;
}

__device__ __forceinline__ void encode_point(const float* __restrict__ x, int p,
                                             const float* __restrict__ table,
                                             float f[8]) {
  float xp0 = (x[3 * p + 0] + 1.f) * 0.5f;
  float xp1 = (x[3 * p + 1] + 1.f) * 0.5f;
  float xp2 = (x[3 * p + 2] + 1.f) * 0.5f;
  // per-level resolutions [2, 5, 12, 32]
  encode_level(xp0 *  2.f, xp1 *  2.f, xp2 *  2.f, table + 0u * TSIZE * 2u, f[0], f[1]);
  encode_level(xp0 *  5.f, xp1 *  5.f, xp2 *  5.f, table + 1u * TSIZE * 2u, f[2], f[3]);
  encode_level(xp0 * 12.f, xp1 * 12.f, xp2 * 12.f, table + 2u * TSIZE * 2u, f[4], f[5]);
  encode_level(xp0 * 32.f, xp1 * 32.f, xp2 * 32.f, table + 3u * TSIZE * 2u, f[6], f[7]);
}

// Scalar fallback MLP for ragged tail (keeps WMMA path full-wave only).
__device__ __forceinline__ void mlp_scalar(const float f[8],
    const float* __restrict__ W0, const float* __restrict__ b0,
    const float* __restrict__ W1, const float* __restrict__ b1,
    const float* __restrict__ W2, const float* __restrict__ b2,
    bool relu_on, float out2[2]) {
  float h0[16];
#pragma unroll
  for (int n2 = 0; n2 < 16; ++n2) {
    float acc = b0[n2];
#pragma unroll
    for (int k = 0; k < 8; ++k) acc = fmaf(f[k], W0[n2 * 8 + k], acc);
    h0[n2] = relu_on ? fmaxf(acc, 0.f) : acc;
  }
  float h1[16];
#pragma unroll
  for (int n2 = 0; n2 < 16; ++n2) {
    float acc = b1[n2];
#pragma unroll
    for (int k = 0; k < 16; ++k) acc = fmaf(h0[k], W1[n2 * 16 + k], acc);
    h1[n2] = relu_on ? fmaxf(acc, 0.f) : acc;
  }
#pragma unroll
  for (int o = 0; o < 2; ++o) {
    float acc = b2[o];
#pragma unroll
    for (int k = 0; k < 16; ++k) acc = fmaf(h1[k], W2[o * 16 + k], acc);
    out2[o] = acc;
  }
}

__global__ __launch_bounds__(256) void ngp_hash_mlp_kernel(
    const float* __restrict__ x, const float* __restrict__ table,
    const float* __restrict__ W0, const float* __restrict__ b0,
    const float* __restrict__ W1, const float* __restrict__ b1,
    const float* __restrict__ W2, const float* __restrict__ b2,
    const int* __restrict__ group, float* __restrict__ out, int n) {
  // Per-wave private LDS staging (no cross-wave sharing -> no barriers).
  __shared__ float featbuf[WPB][32][8];   // [wave][point-in-group][feature]
  __shared__ float hbuf[WPB][16][16];     // [wave][point-in-tile][hidden]

  const int tid  = threadIdx.x;
  const int lane = tid & 31;
  const int wv   = tid >> 5;
  const int ncol = lane & 15;      // WMMA column (N) / A-row (M) owned by lane
  const int hi   = lane >> 4;      // 0 or 1
  const int kb   = hi * 2;         // K-pair base within a K=4 chunk

  const bool relu_on = (group[0] == 1);

  // ---- hoisted per-lane WMMA weight operands ----
  // B (KxN) layout mirrors C/D: lane holds column ncol; VGPR0=B[kb][n], VGPR1=B[kb+1][n].
  // Layer0 GEMM: D[m][n] = sum_k feat[m][k] * W0[n][k]  => B[k][n] = W0[n*8+k]
  v2f B0[2], B1[4];
#pragma unroll
  for (int j = 0; j < 2; ++j) {
    B0[j].x = W0[ncol * 8 + 4 * j + kb];
    B0[j].y = W0[ncol * 8 + 4 * j + kb + 1];
  }
#pragma unroll
  for (int j = 0; j < 4; ++j) {
    B1[j].x = W1[ncol * 16 + 4 * j + kb];
    B1[j].y = W1[ncol * 16 + 4 * j + kb + 1];
  }
  const float bias0 = b0[ncol];
  const float bias1 = b1[ncol];
  // Layer2 done as a per-lane dot: lanes 0-15 -> out feature 0, lanes 16-31 -> 1.
  float w2r[16];
#pragma unroll
  for (int k = 0; k < 16; ++k) w2r[k] = W2[hi * 16 + k];
  const float bias2 = b2[hi];

  const int stride = gridDim.x * blockDim.x;        // points per grid sweep
  const int n_main = n & ~31;                       // full 32-point groups

  for (int base = blockIdx.x * blockDim.x + wv * 32; base + 32 <= n; base += stride) {
    const int p0 = base + lane;                     // one point per lane

    // ---------------- encode phase (VALU + gathers) ----------------
    float f[8];
    encode_point(x, p0, table, f);
#pragma unroll
    for (int k = 0; k < 8; ++k) featbuf[wv][lane][k] = f[k];
    lds_fence_wave();

    // ---------------- WMMA MLP, two 16-point tiles ----------------
#pragma unroll
    for (int t = 0; t < 2; ++t) {
      const int mrow = t * 16 + ncol;               // A-matrix row owned by lane

      // layer 0: [16x8] @ [8x16] as 2x V_WMMA_F32_16X16X4_F32, C preloaded w/ bias
      v8f c;
#pragma unroll
      for (int v = 0; v < 8; ++v) c[v] = bias0;
#pragma unroll
      for (int j = 0; j < 2; ++j) {
        v2f a;
        a.x = featbuf[wv][mrow][4 * j + kb];
        a.y = featbuf[wv][mrow][4 * j + kb + 1];
        c = __builtin_amdgcn_wmma_f32_16x16x4_f32(false, a, false, B0[j],
                                                  (short)0, c, false, false);
      }
      if (relu_on) {
#pragma unroll
        for (int v = 0; v < 8; ++v) c[v] = fmaxf(c[v], 0.f);
      }
      // D layout: VGPR v holds (m = v + 8*hi, n = ncol) -> stage h to LDS
#pragma unroll
      for (int v = 0; v < 8; ++v) hbuf[wv][v + 8 * hi][ncol] = c[v];
      lds_fence_wave();

      // layer 1: [16x16] @ [16x16] as 4x V_WMMA_F32_16X16X4_F32
      v8f d;
#pragma unroll
      for (int v = 0; v < 8; ++v) d[v] = bias1;
#pragma unroll
      for (int j = 0; j < 4; ++j) {
        v2f a;
        a.x = hbuf[wv][ncol][4 * j + kb];
        a.y = hbuf[wv][ncol][4 * j + kb + 1];
        d = __builtin_amdgcn_wmma_f32_16x16x4_f32(false, a, false, B1[j],
                                                  (short)0, d, false, false);
      }
      if (relu_on) {
#pragma unroll
        for (int v = 0; v < 8; ++v) d[v] = fmaxf(d[v], 0.f);
      }
#pragma unroll
      for (int v = 0; v < 8; ++v) hbuf[wv][v + 8 * hi][ncol] = d[v];
      lds_fence_wave();

      // layer 2: 16->2 per-lane dot; lane -> out[base + t*16 + ncol][hi]
      float acc = bias2;
#pragma unroll
      for (int k = 0; k < 16; ++k) acc = fmaf(hbuf[wv][ncol][k], w2r[k], acc);
      out[(base + t * 16 + ncol) * 2 + hi] = acc;
      lds_fence_wave();   // before next tile/iteration reuses hbuf/featbuf
    }
  }

  // ---------------- scalar tail (n not multiple of 32) ----------------
  if (blockIdx.x == 0 && wv == 0) {
    const int tail = n - n_main;
    if (lane < tail) {
      const int p = n_main + lane;
      float f[8], o2[2];
      encode_point(x, p, table, f);
      mlp_scalar(f, W0, b0, W1, b1, W2, b2, relu_on, o2);
      out[p * 2 + 0] = o2[0];
      out[p * 2 + 1] = o2[1];
    }
  }
}

extern "C" void kernel_launch(void* const* d_in, const int* in_sizes, int n_in,
                              void* d_out, int out_size, void* d_ws, size_t ws_size,
                              hipStream_t stream) {
  const float* x     = (const float*)d_in[0];
  const float* table = (const float*)d_in[1];
  const float* W0    = (const float*)d_in[2];
  const float* b0    = (const float*)d_in[3];
  const float* W1    = (const float*)d_in[4];
  const float* b1    = (const float*)d_in[5];
  const float* W2    = (const float*)d_in[6];
  const float* b2    = (const float*)d_in[7];
  const int*   group = (const int*)d_in[8];
  float* out = (float*)d_out;

  const int n = in_sizes[0] / 3;          // N_PTS (x is [N,3])
  const int threads = 256;                // 8 wave32s per block
  const int pts_per_thread = 4;           // grid-stride: amortize weight preload
  int blocks = (n + threads * pts_per_thread - 1) / (threads * pts_per_thread);
  if (blocks < 1) blocks = 1;

  ngp_hash_mlp_kernel<<<blocks, threads, 0, stream>>>(
      x, table, W0, b0, W1, b1, W2, b2, group, out, n);
}